// TopKRouter_78254304133256
// MI455X (gfx1250) — compile-verified
//
#include <hip/hip_runtime.h>
#include <hip/hip_bf16.h>
#include <math.h>

// ---------------------------------------------------------------------------
// Top-2 MoE router for MI455X (gfx1250, wave32).
//   logits[t, e] = dot(x[t, :1024], W[e, :1024]) + b[e]
//   gates = softmax(logits); return top-2 gates + indices.
//
// GEMM on V_WMMA_F32_16X16X4_F32 (exact fp32). Kernel is HBM-bound streaming
// x (134 MB -> ~5.8 us @ 23.3 TB/s); W (256 KB) is staged per-256-k chunk in
// LDS in its NATIVE row-major layout with a 260-float row pitch:
//   - B-fragment reads hit banks (4n + k) % 64 -> fully conflict-free
//   - all inner-loop LDS addresses are base + compile-time constant
//     -> ds_load_b64 with immediate offsets, zero address VALU per step.
// W chunk is staged with the CDNA5 async global->LDS DMA path (ASYNCcnt).
// ---------------------------------------------------------------------------

typedef float v2f __attribute__((ext_vector_type(2)));
typedef float v8f __attribute__((ext_vector_type(8)));

#define IN_FEATURES   1024
#define NUM_EXPERTS   64
#define KC            256               // K-chunk of W staged in LDS
#define LPITCH        260               // padded row pitch (floats): bank = (4n+k)%64
#define WAVES_PER_BLK 8
#define TOKENS_PER_WAVE 16
#define THREADS       (WAVES_PER_BLK * 32)

#if __has_builtin(__builtin_amdgcn_global_load_async_to_lds_b128)
#define HAVE_ASYNC_LDS 1
// Signature (from hipcc diagnostic): param type is
//   'int __attribute__((vector_size(16))) __device__ *'  -> AS1-qualified v4i*
typedef int v4i_ __attribute__((vector_size(16)));
typedef __attribute__((address_space(1))) v4i_ g_v4i;   // global
typedef __attribute__((address_space(3))) v4i_ l_v4i;   // LDS
#endif

// Merge two (top1, top2) candidate pairs; ties -> lower expert index (JAX top_k).
__device__ __forceinline__ void merge2(float& m1, int& i1, float& m2, int& i2,
                                       float o1, int oi1, float o2, int oi2) {
    bool bw  = (o1 > m1) || (o1 == m1 && oi1 < i1);
    float t1 = bw ? o1  : m1;  int ti1 = bw ? oi1 : i1;
    float c  = bw ? m1  : o1;  int ci  = bw ? i1  : oi1;   // loser of firsts
    float d  = bw ? o2  : m2;  int di  = bw ? oi2 : i2;    // second of winner
    bool cw  = (c > d) || (c == d && ci < di);
    m1 = t1; i1 = ti1;
    m2 = cw ? c : d;  i2 = cw ? ci : di;
}

__global__ __launch_bounds__(THREADS)
void topk_router_wmma(const float* __restrict__ x,
                      const float* __restrict__ W,
                      const float* __restrict__ b,
                      float* __restrict__ out, int T) {
    // W chunk, native row-major: element (k, n) at ldsW[n*LPITCH + k].
    __shared__ float ldsW[NUM_EXPERTS * LPITCH];

    const int tid  = threadIdx.x;
    const int wave = tid >> 5;
    const int lane = tid & 31;
    const int half = lane >> 4;       // 0: lanes 0-15, 1: lanes 16-31
    const int n0   = lane & 15;

    const int tileBase = (blockIdx.x * WAVES_PER_BLK + wave) * TOKENS_PER_WAVE;

    // A-fragment source: lane holds row m = n0, K phase 2*half (ISA 16x4 layout).
    const float* xLane = x + (size_t)(tileBase + n0) * IN_FEATURES + 2 * half;
    // B-fragment base: expert n0 (+nb*16 via immediate), K phase 2*half.
    const float* ldsB  = ldsW + n0 * LPITCH + 2 * half;

    v8f acc[4] = {};                  // 4 N-blocks of 16 -> 64 experts

    for (int kc = 0; kc < IN_FEATURES; kc += KC) {
        // ---- stage W[:, kc:kc+KC] into LDS (no transpose needed) ----
        #pragma unroll
        for (int i = 0; i < (KC * NUM_EXPERTS) / (4 * THREADS); ++i) {  // 16 iters
            int idx4 = i * THREADS + tid;         // float4 slots, 0..4095
            int n    = idx4 >> 6;                 // 64 float4 per expert row
            int k    = (idx4 & 63) << 2;          // 0..252 within chunk
            const float* gsrc = W + (size_t)n * IN_FEATURES + kc + k;
            float*       ldst = ldsW + n * LPITCH + k;
#if HAVE_ASYNC_LDS
            // CDNA5 async DMA: global -> LDS without a VGPR round-trip (ASYNCcnt).
            __builtin_amdgcn_global_load_async_to_lds_b128(
                (g_v4i*)gsrc, (l_v4i*)ldst, 0, 0);
#else
            *(float4*)ldst = *(const float4*)gsrc;
#endif
        }
#if HAVE_ASYNC_LDS
#if __has_builtin(__builtin_amdgcn_s_wait_asynccnt)
        __builtin_amdgcn_s_wait_asynccnt(0);
#else
        asm volatile("s_wait_asynccnt 0x0" ::: "memory");
#endif
#endif
        __syncthreads();

        // ---- WMMA GEMM over this K chunk ----
        const float* xk = xLane + kc;
        #pragma unroll 4
        for (int kk = 0; kk < KC; kk += 8) {
            if ((kk & 63) == 0)                   // keep the x stream ahead in GL2
                __builtin_prefetch(xk + kk + KC, 0, 1);

            // Grouped loads (all immediate-offset) ...
            v2f a0 = *(const v2f*)(xk + kk);
            v2f a1 = *(const v2f*)(xk + kk + 4);
            v2f bf0[4], bf1[4];
            #pragma unroll
            for (int nb = 0; nb < 4; ++nb)
                bf0[nb] = *(const v2f*)(ldsB + nb * (16 * LPITCH) + kk);
            #pragma unroll
            for (int nb = 0; nb < 4; ++nb)
                bf1[nb] = *(const v2f*)(ldsB + nb * (16 * LPITCH) + kk + 4);

            // ... then grouped MACs, so loads clause & overlap the WMMAs.
            #pragma unroll
            for (int nb = 0; nb < 4; ++nb)
                acc[nb] = __builtin_amdgcn_wmma_f32_16x16x4_f32(
                    false, a0, false, bf0[nb], (short)0, acc[nb], false, false);
            #pragma unroll
            for (int nb = 0; nb < 4; ++nb)
                acc[nb] = __builtin_amdgcn_wmma_f32_16x16x4_f32(
                    false, a1, false, bf1[nb], (short)0, acc[nb], false, false);
        }
        __syncthreads();
    }

    // ---- bias + softmax + top-2 ----
    float bv[4];
    #pragma unroll
    for (int nb = 0; nb < 4; ++nb) bv[nb] = b[nb * 16 + n0];

    const int T2 = T * 2;

    #pragma unroll
    for (int r = 0; r < 8; ++r) {
        // This (lane, r) holds logit for row M = r + 8*half, expert n = nb*16 + n0.
        float v[4];
        #pragma unroll
        for (int nb = 0; nb < 4; ++nb) v[nb] = acc[nb][r] + bv[nb];

        // local top-2 over the 4 experts this lane owns
        float m1 = v[0]; int i1 = n0;
        float m2 = -INFINITY; int i2 = 0x7fffffff;
        #pragma unroll
        for (int nb = 1; nb < 4; ++nb) {
            int n = nb * 16 + n0;
            if (v[nb] > m1)      { m2 = m1; i2 = i1; m1 = v[nb]; i1 = n; }
            else if (v[nb] > m2) { m2 = v[nb]; i2 = n; }
        }

        // reduce across the 16 lanes of this half
        #pragma unroll
        for (int mask = 1; mask <= 8; mask <<= 1) {
            float o1 = __shfl_xor(m1, mask, 16);
            int  oi1 = __shfl_xor(i1, mask, 16);
            float o2 = __shfl_xor(m2, mask, 16);
            int  oi2 = __shfl_xor(i2, mask, 16);
            merge2(m1, i1, m2, i2, o1, oi1, o2, oi2);
        }

        // softmax denominator: sum over all 64 experts of exp(l - max)
        float z = 0.0f;
        #pragma unroll
        for (int nb = 0; nb < 4; ++nb) z += __expf(v[nb] - m1);
        #pragma unroll
        for (int mask = 1; mask <= 8; mask <<= 1)
            z += __shfl_xor(z, mask, 16);

        if (n0 == 0) {
            int token = tileBase + r + 8 * half;
            float inv = 1.0f / z;
            out[token * 2 + 0] = inv;                    // exp(m1-m1)/Z
            out[token * 2 + 1] = __expf(m2 - m1) * inv;  // exp(m2-m1)/Z
            out[T2 + token * 2 + 0] = (float)i1;         // indices, as output dtype
            out[T2 + token * 2 + 1] = (float)i2;
        }
    }
}

extern "C" void kernel_launch(void* const* d_in, const int* in_sizes, int n_in,
                              void* d_out, int out_size, void* d_ws, size_t ws_size,
                              hipStream_t stream) {
    (void)n_in; (void)out_size; (void)d_ws; (void)ws_size;
    const float* x = (const float*)d_in[0];   // [8*4096, 1024]
    const float* W = (const float*)d_in[1];   // [64, 1024]
    const float* b = (const float*)d_in[2];   // [64]
    float* out = (float*)d_out;               // [2T gates | 2T indices]

    const int T = in_sizes[0] / IN_FEATURES;                   // 32768 tokens
    const int blocks = T / (WAVES_PER_BLK * TOKENS_PER_WAVE);  // 256

    topk_router_wmma<<<blocks, THREADS, 0, stream>>>(x, W, b, out, T);
}